// BatchAllTtripletLoss_multi_module_version_55671366091650
// MI455X (gfx1250) — compile-verified
//
#include <hip/hip_runtime.h>

#define B_     192
#define D_     512
#define N2     384            // 2*B
#define EPSF   1e-8f
#define MARGIN 0.1f

typedef float v2f __attribute__((ext_vector_type(2)));
typedef float v8f __attribute__((ext_vector_type(8)));

// ---------------------------------------------------------------------------
// Kernel 1: L2-normalize rows of output1/output2 into emb (384 x 512)
// ---------------------------------------------------------------------------
__global__ void normalize_rows_kernel(const float* __restrict__ o1,
                                      const float* __restrict__ o2,
                                      float* __restrict__ emb) {
  int row = blockIdx.x;                       // 0..383
  const float* src = (row < B_) ? (o1 + row * D_) : (o2 + (row - B_) * D_);
  float* dst = emb + row * D_;
  int t = threadIdx.x;                        // 256 threads, 2 elems each
  float x0 = src[t];
  float x1 = src[t + 256];

  __shared__ float red[256];
  red[t] = x0 * x0 + x1 * x1;
  __syncthreads();
  for (int st = 128; st > 0; st >>= 1) {
    if (t < st) red[t] += red[t + st];
    __syncthreads();
  }
  __shared__ float s_inv;
  if (t == 0) {
    float n = sqrtf(red[0]);
    n = fmaxf(n, 1e-12f);                     // clip like reference
    s_inv = 1.0f / n;
  }
  __syncthreads();
  float iv = s_inv;
  dst[t]       = x0 * iv;
  dst[t + 256] = x1 * iv;
}

// ---------------------------------------------------------------------------
// Kernel 2: Gram matrix G = emb @ emb^T  (384x384), f32 WMMA 16x16x4.
// One wave per 16x16 output tile; 24x24 = 576 tiles = 144 blocks x 4 waves.
// A 16x4 f32 layout (ISA 7.12.2): lanes 0-15 -> M=lane, v0=K+0, v1=K+1;
//                                 lanes 16-31 -> M=lane-16, v0=K+2, v1=K+3.
// B 4x16 f32 (rows striped across lanes): lane n (0-15): v0=B[K+0][n]=emb[n][K+0],
//   v1=emb[n][K+1]; lanes 16-31: K+2/K+3.  Same addressing pattern as A.
// C/D 16x16 f32: VGPR v -> M = v + 8*(lane/16), N = lane%16.
// ---------------------------------------------------------------------------
__global__ void gram_wmma_kernel(const float* __restrict__ emb,
                                 float* __restrict__ G) {
  int wave  = blockIdx.x * 4 + (threadIdx.x >> 5);   // exactly 576 waves total
  int tileM = wave / 24;
  int tileN = wave % 24;
  int lane  = threadIdx.x & 31;
  int half  = lane >> 4;                             // 0 or 1
  int l16   = lane & 15;

  const float* arow = emb + (tileM * 16 + l16) * D_ + 2 * half;
  const float* brow = emb + (tileN * 16 + l16) * D_ + 2 * half;

  v8f acc = {};
#pragma unroll 8
  for (int k = 0; k < D_; k += 4) {
    v2f a = *(const v2f*)(arow + k);
    v2f b = *(const v2f*)(brow + k);
    acc = __builtin_amdgcn_wmma_f32_16x16x4_f32(
        /*neg_a=*/false, a, /*neg_b=*/false, b,
        /*c_mod=*/(short)0, acc, /*reuse_a=*/false, /*reuse_b=*/false);
  }

  int col   = tileN * 16 + l16;
  int rbase = tileM * 16 + 8 * half;
#pragma unroll
  for (int v = 0; v < 8; ++v) {
    G[(rbase + v) * N2 + col] = acc[v];
  }
}

// ---------------------------------------------------------------------------
// Kernel 3: collapsed triplet loss. Block i = anchor. j = (i+192)%384 is the
// unique positive; k ranges over labels[k] != labels[i]. Reference semantics:
// sq = diag(G); d2 = relu(sq_k - 2G + sq_i); zmask = (d2==0);
// d = sqrt(d2 + zmask*eps) * (1 - zmask)  ==>  d = (d2==0) ? 0 : sqrt(d2).
// ---------------------------------------------------------------------------
__global__ void triplet_loss_kernel(const float* __restrict__ G,
                                    const float* __restrict__ weight,
                                    float* __restrict__ part) {
  int i = blockIdx.x;                         // 0..383
  int t = threadIdx.x;                        // 128 threads
  int imod = (i < B_) ? i : (i - B_);

  __shared__ float s_dij, s_sqi;
  if (t == 0) {
    int j = (i + B_) % N2;
    float sqi = G[i * N2 + i];
    float sqj = G[j * N2 + j];
    float d2  = fmaxf(sqj - 2.0f * G[i * N2 + j] + sqi, 0.0f);
    s_dij = (d2 == 0.0f) ? 0.0f : sqrtf(d2);
    s_sqi = sqi;
  }
  __syncthreads();
  float dij = s_dij;
  float sqi = s_sqi;

  float sum = 0.0f, cnt = 0.0f;
  for (int k = t; k < N2; k += 128) {
    int kmod = (k < B_) ? k : (k - B_);
    if (kmod == imod) continue;               // excludes k==i and k==i+-192
    float sqk = G[k * N2 + k];
    float d2  = fmaxf(sqk - 2.0f * G[i * N2 + k] + sqi, 0.0f);
    float dik = (d2 == 0.0f) ? 0.0f : sqrtf(d2);
    float tl  = dij - dik + MARGIN;
    if (tl > 0.0f) {
      float c = tl * weight[imod * B_ + kmod];
      sum += c;
      if (c > EPSF) cnt += 1.0f;
    }
  }

  __shared__ float rs[128], rc[128];
  rs[t] = sum; rc[t] = cnt;
  __syncthreads();
  for (int st = 64; st > 0; st >>= 1) {
    if (t < st) { rs[t] += rs[t + st]; rc[t] += rc[t + st]; }
    __syncthreads();
  }
  if (t == 0) { part[i] = rs[0]; part[N2 + i] = rc[0]; }
}

// ---------------------------------------------------------------------------
// Kernel 4: final reduction over 384 per-anchor partials -> scalar loss.
// ---------------------------------------------------------------------------
__global__ void finalize_kernel(const float* __restrict__ part,
                                float* __restrict__ out) {
  int t = threadIdx.x;                        // 128 threads
  float s = 0.0f, c = 0.0f;
  for (int i = t; i < N2; i += 128) { s += part[i]; c += part[N2 + i]; }
  __shared__ float rs[128], rc[128];
  rs[t] = s; rc[t] = c;
  __syncthreads();
  for (int st = 64; st > 0; st >>= 1) {
    if (t < st) { rs[t] += rs[t + st]; rc[t] += rc[t + st]; }
    __syncthreads();
  }
  if (t == 0) out[0] = rs[0] / (rc[0] + EPSF);
}

// ---------------------------------------------------------------------------
extern "C" void kernel_launch(void* const* d_in, const int* in_sizes, int n_in,
                              void* d_out, int out_size, void* d_ws, size_t ws_size,
                              hipStream_t stream) {
  const float* o1 = (const float*)d_in[0];    // (192, 512) f32
  const float* o2 = (const float*)d_in[1];    // (192, 512) f32
  const float* w  = (const float*)d_in[2];    // (192, 192) f32

  float* emb  = (float*)d_ws;                 // 384*512 floats
  float* G    = emb + N2 * D_;                // 384*384 floats
  float* part = G + N2 * N2;                  // 2*384 floats

  normalize_rows_kernel<<<N2, 256, 0, stream>>>(o1, o2, emb);
  gram_wmma_kernel<<<144, 128, 0, stream>>>(emb, G);
  triplet_loss_kernel<<<N2, 128, 0, stream>>>(G, w, part);
  finalize_kernel<<<1, 128, 0, stream>>>(part, (float*)d_out);
}